// GCN_designed_29300266893374
// MI455X (gfx1250) — compile-verified
//
#include <hip/hip_runtime.h>
#include <hip/hip_bf16.h>
#include <math.h>

typedef __attribute__((ext_vector_type(2))) float v2f;
typedef __attribute__((ext_vector_type(4))) float v4f;
typedef __attribute__((ext_vector_type(8))) float v8f;
typedef __attribute__((ext_vector_type(4))) int   v4i;

#define NBINS 256

// ---------------- zero accumulators ----------------
__global__ void k_zero(float* __restrict__ acc, float* __restrict__ hg, int n) {
    int i = blockIdx.x * blockDim.x + threadIdx.x;
    if (i < n)     acc[i] = 0.0f;
    if (i < NBINS) hg[i]  = 0.0f;
}

// ---------------- layer-0 projection via V_WMMA_F32_16X16X4_F32 ----------------
// For a tile of 16 nodes: D = A(16x8 node feats) x B(8x2), with B column 0 =
// W0n (neighbor weight) and column 1 = W0s (self weight). K=8 is split into
// two K=4 WMMA steps accumulating into C.
// A layout (16x4 f32): lane m (0..15): v0=K0,v1=K1 ; lane m+16: v0=K2,v1=K3.
// B layout (4x16 f32): lane n = column; v0=K0/K2, v1=K1/K3 by lane-half.
// D layout (16x16 f32): lane = column n; c[r] = D[(laneHalf*8)+r][n].
__global__ void k_proj0(const float* __restrict__ x,
                        const float* __restrict__ W0n,
                        const float* __restrict__ W0s,
                        float* __restrict__ xwn,   // x @ W0n  (to be scattered)
                        float* __restrict__ h,     // x @ W0s  (self term, pre-act)
                        int n_nodes) {
    const int lane = threadIdx.x & 31;
    const int wave = threadIdx.x >> 5;
    const int tile = blockIdx.x * (blockDim.x >> 5) + wave;
    const int base = tile * 16;
    if (base >= n_nodes) return;            // wave-uniform: EXEC stays all-1s

    const int m  = lane & 15;               // node-in-tile (A) / column (B,D)
    const int kb = (lane >> 4) * 2;         // K interleave per lane-half: 0 or 2

    int row = base + m;
    if (row >= n_nodes) row = n_nodes - 1;  // clamp for partial tiles
    const float* xr = x + (size_t)row * 8;

    v2f a0, a1, b0v, b1v;
    a0.x = xr[kb + 0]; a0.y = xr[kb + 1];   // K chunk 0..3
    a1.x = xr[kb + 4]; a1.y = xr[kb + 5];   // K chunk 4..7

    // B: column 0 = W0n, column 1 = W0s, all other columns zero
    const float* w = (m == 0) ? W0n : ((m == 1) ? W0s : (const float*)0);
    if (w) {
        b0v.x = w[kb + 0]; b0v.y = w[kb + 1];
        b1v.x = w[kb + 4]; b1v.y = w[kb + 5];
    } else {
        b0v.x = 0.0f; b0v.y = 0.0f;
        b1v.x = 0.0f; b1v.y = 0.0f;
    }

    v8f c = {0.0f, 0.0f, 0.0f, 0.0f, 0.0f, 0.0f, 0.0f, 0.0f};
    c = __builtin_amdgcn_wmma_f32_16x16x4_f32(false, a0, false, b0v, (short)0, c, false, false);
    c = __builtin_amdgcn_wmma_f32_16x16x4_f32(false, a1, false, b1v, (short)0, c, false, false);

    const int row0 = base + (lane >> 4) * 8;   // D rows this lane-half holds
    if (base + 16 <= n_nodes) {
        // full tile: two 128-bit stores per producing lane (global_store_b128)
        v4f lo, hi;
        lo.x = c[0]; lo.y = c[1]; lo.z = c[2]; lo.w = c[3];
        hi.x = c[4]; hi.y = c[5]; hi.z = c[6]; hi.w = c[7];
        if (m == 0) {
            *(v4f*)(xwn + row0)     = lo;
            *(v4f*)(xwn + row0 + 4) = hi;
        } else if (m == 1) {
            *(v4f*)(h + row0)     = lo;
            *(v4f*)(h + row0 + 4) = hi;
        }
    } else {
        if (m == 0) {
#pragma unroll
            for (int r = 0; r < 8; ++r)
                if (row0 + r < n_nodes) xwn[row0 + r] = c[r];
        } else if (m == 1) {
#pragma unroll
            for (int r = 0; r < 8; ++r)
                if (row0 + r < n_nodes) h[row0 + r] = c[r];
        }
    }
}

// ---------------- edge scatter: acc[dst] += val[src], 4 edges/thread ----------------
// Edge lists (128 MB) and node arrays (4 MB) fit in the 192 MB L2, so both the
// gathers and the GLOBAL_ATOMIC_ADD_F32 (no return -> STOREcnt only) hit L2.
// int4 index loads turn 8x b32 loads into 2x b128 per 4 edges.
__global__ void k_scatter(const int* __restrict__ src, const int* __restrict__ dst,
                          const float* __restrict__ val, float* __restrict__ acc,
                          int ne) {
    int base = (blockIdx.x * blockDim.x + threadIdx.x) * 4;
    if (base + 4 <= ne) {
        v4i s = *(const v4i*)(src + base);
        v4i d = *(const v4i*)(dst + base);
        float v0 = val[s.x], v1 = val[s.y], v2 = val[s.z], v3 = val[s.w];
        atomicAdd(&acc[d.x], v0);
        atomicAdd(&acc[d.y], v1);
        atomicAdd(&acc[d.z], v2);
        atomicAdd(&acc[d.w], v3);
    } else {
        for (int e = base; e < ne; ++e)
            atomicAdd(&acc[dst[e]], val[src[e]]);
    }
}

// ---------------- node updates (DIM == 1, so weights are scalars) ----------------
__global__ void k_update0(float* __restrict__ h, float* __restrict__ acc,
                          const float* __restrict__ b0, int n) {
    int i = blockIdx.x * blockDim.x + threadIdx.x;
    if (i >= n) return;
    float v = acc[i] + b0[0] + h[i];   // h currently holds x@W0s
    h[i]   = fmaxf(v, 0.0f);
    acc[i] = 0.0f;                     // re-arm accumulator for next layer
}

__global__ void k_updatek(float* __restrict__ h, float* __restrict__ acc,
                          const float* __restrict__ Wn, const float* __restrict__ b,
                          const float* __restrict__ Ws, int layer, int n) {
    int i = blockIdx.x * blockDim.x + threadIdx.x;
    if (i >= n) return;
    float wn = Wn[layer], bb = b[layer], ws = Ws[layer];
    float v = wn * acc[i] + bb + ws * h[i];
    h[i]   = fmaxf(v, 0.0f);
    acc[i] = 0.0f;
}

// ---------------- graph readout with LDS bins (graph_ids is sorted) ----------------
__global__ void k_readout(const float* __restrict__ h, const int* __restrict__ gid,
                          float* __restrict__ hg, int n) {
    __shared__ float bins[NBINS];
    int t = threadIdx.x;
    bins[t] = 0.0f;
    __syncthreads();
    for (int i = blockIdx.x * blockDim.x + t; i < n; i += gridDim.x * blockDim.x)
        atomicAdd(&bins[gid[i]], h[i]);          // ds_add_f32; sorted ids -> few bins/block
    __syncthreads();
    float v = bins[t];
    if (v != 0.0f) atomicAdd(&hg[t], v);         // ~1-2 global atomics per block
}

// ---------------- MLP head + log_softmax (one thread per graph) ----------------
__device__ inline float stable_sigmoid(float z) {
    if (z >= 0.0f) { float e = __expf(-z); return 1.0f / (1.0f + e); }
    float e = __expf(z); return e / (1.0f + e);
}

__global__ void k_head(const float* __restrict__ hg,
                       const float* __restrict__ fc1_w, const float* __restrict__ fc1_b,
                       const float* __restrict__ out_w, const float* __restrict__ out_b,
                       float* __restrict__ out, int G) {
    int g = blockIdx.x * blockDim.x + threadIdx.x;
    if (g >= G) return;
    float t = hg[g];
    float a[8];
#pragma unroll
    for (int j = 0; j < 8; ++j)
        a[j] = stable_sigmoid((t * fc1_w[j] + fc1_b[j]) * 1000.0f);
    float o[4];
#pragma unroll
    for (int c = 0; c < 4; ++c) {
        float s = out_b[c];
#pragma unroll
        for (int j = 0; j < 8; ++j) s += a[j] * out_w[j * 4 + c];
        o[c] = fmaxf(s, 0.0f);
    }
    float m = fmaxf(fmaxf(o[0], o[1]), fmaxf(o[2], o[3]));
    float se = 0.0f;
#pragma unroll
    for (int c = 0; c < 4; ++c) se += __expf(o[c] - m);
    float lse = m + __logf(se);
#pragma unroll
    for (int c = 0; c < 4; ++c) out[(size_t)g * 4 + c] = o[c] - lse;
}

extern "C" void kernel_launch(void* const* d_in, const int* in_sizes, int n_in,
                              void* d_out, int out_size, void* d_ws, size_t ws_size,
                              hipStream_t stream) {
    const float* x     = (const float*)d_in[0];
    const int*   src   = (const int*)  d_in[1];
    const int*   dst   = (const int*)  d_in[2];
    const int*   gid   = (const int*)  d_in[3];
    // d_in[4] = num_graphs (device scalar; G derived from out_size instead)
    const float* W0n   = (const float*)d_in[5];
    const float* b0    = (const float*)d_in[6];
    const float* W0s   = (const float*)d_in[7];
    const float* Wkn   = (const float*)d_in[8];
    const float* bk    = (const float*)d_in[9];
    const float* Wks   = (const float*)d_in[10];
    const float* fc1_w = (const float*)d_in[11];
    const float* fc1_b = (const float*)d_in[12];
    const float* out_w = (const float*)d_in[13];
    const float* out_b = (const float*)d_in[14];
    float* out = (float*)d_out;

    const int nN = in_sizes[3];     // 1,000,000 nodes
    const int nE = in_sizes[1];     // 16,000,000 edges
    const int G  = out_size / 4;    // 256 graphs

    // workspace: acc[N] | h[N] | xwn[N] | hg[256]   (~12 MB + 1 KB)
    float* acc = (float*)d_ws;
    float* h   = acc + nN;
    float* xwn = h + nN;
    float* hg  = xwn + nN;

    const int TB  = 256;
    const int nbN = (nN + TB - 1) / TB;
    const int nbE = (nE + TB * 4 - 1) / (TB * 4);   // 4 edges per thread

    k_zero<<<nbN, TB, 0, stream>>>(acc, hg, nN);

    // layer 0: WMMA projection, scatter x@W0n, fuse bias+self+relu
    const int tiles  = (nN + 15) / 16;
    const int nbProj = (tiles + 7) / 8;             // 8 wave32s per block
    k_proj0<<<nbProj, 256, 0, stream>>>(x, W0n, W0s, xwn, h, nN);
    k_scatter<<<nbE, TB, 0, stream>>>(src, dst, xwn, acc, nE);
    k_update0<<<nbN, TB, 0, stream>>>(h, acc, b0, nN);

    // layers 1..3: scalar hidden state; scatter raw h, scale in update
    for (int l = 0; l < 3; ++l) {
        k_scatter<<<nbE, TB, 0, stream>>>(src, dst, h, acc, nE);
        k_updatek<<<nbN, TB, 0, stream>>>(h, acc, Wkn, bk, Wks, l, nN);
    }

    k_readout<<<2048, NBINS, 0, stream>>>(h, gid, hg, nN);
    k_head<<<(G + 255) / 256, 256, 0, stream>>>(hg, fc1_w, fc1_b, out_w, out_b, out, G);
}